// QonvLayer_58523224375629
// MI455X (gfx1250) — compile-verified
//
#include <hip/hip_runtime.h>

// ============================================================================
// Compile-time replication of numpy.random.RandomState(1234) build_ops():
// MT19937 init_genrand seeding, legacy random_double (2x u32),
// legacy randint via masked-rejection on next_uint64 (hi<<32|lo),
// legacy shuffle via 32-bit random_interval masked rejection.
// ============================================================================
namespace ct {

struct MT { unsigned mt[624]; int idx; };

constexpr void mt_seed(MT &s, unsigned seed) {
  s.mt[0] = seed;
  for (int i = 1; i < 624; ++i)
    s.mt[i] = 1812433253u * (s.mt[i - 1] ^ (s.mt[i - 1] >> 30)) + (unsigned)i;
  s.idx = 624;
}

constexpr unsigned mt_next(MT &s) {
  if (s.idx >= 624) {
    for (int i = 0; i < 624; ++i) {
      unsigned y = (s.mt[i] & 0x80000000u) | (s.mt[(i + 1) % 624] & 0x7fffffffu);
      unsigned nx = s.mt[(i + 397) % 624] ^ (y >> 1);
      if (y & 1u) nx ^= 0x9908b0dfu;
      s.mt[i] = nx;
    }
    s.idx = 0;
  }
  unsigned y = s.mt[s.idx++];
  y ^= y >> 11;
  y ^= (y << 7) & 0x9d2c5680u;
  y ^= (y << 15) & 0xefc60000u;
  y ^= y >> 18;
  return y;
}

constexpr double mt_double(MT &s) {
  unsigned a = mt_next(s) >> 5;
  unsigned b = mt_next(s) >> 6;
  return (a * 67108864.0 + b) / 9007199254740992.0;
}

constexpr unsigned long long mt_next64(MT &s) {
  unsigned long long hi = mt_next(s);
  unsigned long long lo = mt_next(s);
  return (hi << 32) | lo;
}

// legacy randint(low, high): inclusive range size rng, masked rejection on u64
constexpr unsigned long long bounded64(MT &s, unsigned long long rng) {
  unsigned long long mask = rng;
  mask |= mask >> 1;  mask |= mask >> 2;  mask |= mask >> 4;
  mask |= mask >> 8;  mask |= mask >> 16; mask |= mask >> 32;
  unsigned long long v = mt_next64(s) & mask;
  while (v > rng) v = mt_next64(s) & mask;
  return v;
}

// legacy random_interval (used by shuffle): 32-bit draws when max <= 0xffffffff
constexpr unsigned interval32(MT &s, unsigned mx) {
  if (mx == 0u) return 0u;
  unsigned mask = mx;
  mask |= mask >> 1; mask |= mask >> 2; mask |= mask >> 4;
  mask |= mask >> 8; mask |= mask >> 16;
  unsigned v = mt_next(s) & mask;
  while (v > mx) v = mt_next(s) & mask;
  return v;
}

constexpr int MAX_OPS = 96;
struct OpsTable {
  int n;
  int type[MAX_OPS];  // 0 = rotation, 1 = cnot
  int a[MAX_OPS];     // rot: kind 0=RX 1=RY 2=RZ ; cnot: control wire
  int b[MAX_OPS];     // rot: wire               ; cnot: target wire
  int widx[MAX_OPS];  // rot: flat weight index l*8+i
};

constexpr OpsTable build_ops() {
  OpsTable T{};
  MT s{};
  mt_seed(s, 1234u);
  for (int l = 0; l < 4; ++l) {
    int i = 0;
    while (i < 8) {
      double r = mt_double(s);
      if (r > 0.3) {
        int kind = (int)bounded64(s, 2ull);  // choice(['RX','RY','RZ'])
        int w    = (int)bounded64(s, 3ull);  // randint(4)
        if (T.n < MAX_OPS) {
          T.type[T.n] = 0; T.a[T.n] = kind; T.b[T.n] = w; T.widx[T.n] = l * 8 + i;
          T.n++;
        }
        ++i;
      } else {
        // choice(4, size=2, replace=False) -> permutation(4)[:2] via shuffle
        int arr[4] = {0, 1, 2, 3};
        for (int k = 3; k >= 1; --k) {
          int j = (int)interval32(s, (unsigned)k);
          int t = arr[k]; arr[k] = arr[j]; arr[j] = t;
        }
        if (T.n < MAX_OPS) {
          T.type[T.n] = 1; T.a[T.n] = arr[0]; T.b[T.n] = arr[1]; T.widx[T.n] = 0;
          T.n++;
        }
      }
    }
  }
  return T;
}

}  // namespace ct

__constant__ ct::OpsTable c_ops = ct::build_ops();

// ============================================================================
// Kernel 1: build the fixed 16x16 complex unitary U(weights) on device.
// Wire w maps to state-index bit (3-w). One thread per column.
// ws[0..255]   = Re(U)[row][col], ws[256..511] = Im(U)[row][col].
// Runs once per launch with 16 threads -> keep full-accuracy libm trig here.
// ============================================================================
__global__ void build_unitary_kernel(const float* __restrict__ weights,
                                     float* __restrict__ ws) {
  __shared__ float Ure[16][16];
  __shared__ float Uim[16][16];
  const int c = threadIdx.x;
  if (c >= 16) return;
  for (int r = 0; r < 16; ++r) { Ure[r][c] = (r == c) ? 1.f : 0.f; Uim[r][c] = 0.f; }

  const int nops = c_ops.n;
  for (int o = 0; o < nops; ++o) {
    if (c_ops.type[o] == 0) {
      const int kind = c_ops.a[o];
      const int w    = c_ops.b[o];
      const float th = 0.5f * weights[c_ops.widx[o]];
      const float ch = cosf(th), sh = sinf(th);
      float g00r = ch, g00i = 0.f, g01r = 0.f, g01i = 0.f;
      float g10r = 0.f, g10i = 0.f, g11r = ch, g11i = 0.f;
      if (kind == 0) {            // RX
        g01i = -sh; g10i = -sh;
      } else if (kind == 1) {     // RY
        g01r = -sh; g10r = sh;
      } else {                    // RZ: diag(e^{-i th}, e^{+i th})
        g00i = -sh; g11i = sh;
      }
      const int bit = 1 << (3 - w);
      for (int r0 = 0; r0 < 16; ++r0) {
        if (r0 & bit) continue;
        const int r1 = r0 | bit;
        const float x0r = Ure[r0][c], x0i = Uim[r0][c];
        const float x1r = Ure[r1][c], x1i = Uim[r1][c];
        Ure[r0][c] = g00r * x0r - g00i * x0i + g01r * x1r - g01i * x1i;
        Uim[r0][c] = g00r * x0i + g00i * x0r + g01r * x1i + g01i * x1r;
        Ure[r1][c] = g10r * x0r - g10i * x0i + g11r * x1r - g11i * x1i;
        Uim[r1][c] = g10r * x0i + g10i * x0r + g11r * x1i + g11i * x1r;
      }
    } else {  // CNOT: rows with control bit set swap target-bit pair
      const int cb = 1 << (3 - c_ops.a[o]);
      const int tb = 1 << (3 - c_ops.b[o]);
      for (int r = 0; r < 16; ++r) {
        if ((r & cb) && !(r & tb)) {
          const int r2 = r | tb;
          float t = Ure[r][c]; Ure[r][c] = Ure[r2][c]; Ure[r2][c] = t;
          t = Uim[r][c]; Uim[r][c] = Uim[r2][c]; Uim[r2][c] = t;
        }
      }
    }
  }
  for (int r = 0; r < 16; ++r) {
    ws[r * 16 + c]       = Ure[r][c];
    ws[256 + r * 16 + c] = Uim[r][c];
  }
}

// ============================================================================
// Kernel 2: batched evaluation, one wave per 16 patches.
// Phi = U * Psi as chained V_WMMA_F32_16X16X4_F32 (K=16 in 4 steps, Re + Im).
// D layout: lane l holds column N=l%16 (its own patch), rows M=r+8*(l/16).
// Trig via hardware v_sin/v_cos (args in [0, pi/2] -> fast path is exact-ish).
// ============================================================================
typedef __attribute__((ext_vector_type(2))) float v2f;
typedef __attribute__((ext_vector_type(8))) float v8f;

__global__ void qonv_wmma_kernel(const float* __restrict__ img,
                                 const float* __restrict__ ws,
                                 float* __restrict__ out) {
  const int lane = threadIdx.x & 31;
  const int tile = (blockIdx.x * blockDim.x + threadIdx.x) >> 5;  // 0..16383
  const int col  = lane & 15;   // patch within tile == B column N == A row M
  const int half = lane >> 4;   // selects K pair {0,1} vs {2,3} within chunk

  const int patch = tile * 16 + col;
  const int b = patch >> 14;            // 128*128 patches per image
  const int rem = patch & 16383;
  const int m = rem >> 7;
  const int n = rem & 127;

  // double-stride sampling: pixels (2m,2n),(2m,2n+1),(2m+1,2n),(2m+1,2n+1)
  const float* gp = img + (((b << 9) + (m << 1)) << 9) + (n << 1);
  float c0, s0, c1, s1, c2, s2, c3, s3;
  __sincosf(0.5f * gp[0],   &s0, &c0);
  __sincosf(0.5f * gp[1],   &s1, &c1);
  __sincosf(0.5f * gp[512], &s2, &c2);
  __sincosf(0.5f * gp[513], &s3, &c3);

  // product-state amplitude psi_k (real); wire w <-> bit (3-w)
  auto psi = [&](int k) -> float {
    float r = (k & 8) ? s0 : c0;
    r *= (k & 4) ? s1 : c1;
    r *= (k & 2) ? s2 : c2;
    r *= (k & 1) ? s3 : c3;
    return r;
  };

  const int kb = half << 1;
  v2f B[4], Ar[4], Ai[4];
#pragma unroll
  for (int q = 0; q < 4; ++q) {
    const int k = (q << 2) + kb;  // global K index of this lane's pair
    B[q]  = v2f{psi(k), psi(k + 1)};
    Ar[q] = v2f{ws[col * 16 + k],       ws[col * 16 + k + 1]};
    Ai[q] = v2f{ws[256 + col * 16 + k], ws[256 + col * 16 + k + 1]};
  }

  v8f Dr = {0.f, 0.f, 0.f, 0.f, 0.f, 0.f, 0.f, 0.f};
  v8f Di = {0.f, 0.f, 0.f, 0.f, 0.f, 0.f, 0.f, 0.f};
#pragma unroll
  for (int q = 0; q < 4; ++q) {
    Dr = __builtin_amdgcn_wmma_f32_16x16x4_f32(false, Ar[q], false, B[q],
                                               (short)0, Dr, false, false);
    Di = __builtin_amdgcn_wmma_f32_16x16x4_f32(false, Ai[q], false, B[q],
                                               (short)0, Di, false, false);
  }

  // probs and signed sums: exp_j = sum_M sign(bit(3-j) of M) * |phi_M|^2
  float e0 = 0.f, e1 = 0.f, e2 = 0.f, e3 = 0.f;
#pragma unroll
  for (int r = 0; r < 8; ++r) {
    const float p = Dr[r] * Dr[r] + Di[r] * Di[r];
    const int M = r + (half << 3);
    e0 += (M & 8) ? -p : p;
    e1 += (M & 4) ? -p : p;
    e2 += (M & 2) ? -p : p;
    e3 += (M & 1) ? -p : p;
  }
  // combine row halves (lanes l and l+16 hold the same patch)
  e0 += __shfl_xor(e0, 16, 32);
  e1 += __shfl_xor(e1, 16, 32);
  e2 += __shfl_xor(e2, 16, 32);
  e3 += __shfl_xor(e3, 16, 32);

  if (lane < 16) {
    float4 o;
    o.x = e0; o.y = e1; o.z = e2; o.w = e3;
    const int oidx = ((((b << 8) + m) << 8) + n) << 2;  // (16,256,256,4)
    *reinterpret_cast<float4*>(out + oidx) = o;
  }
}

// Zero-fill only the part of (16,256,256,4) that the compute kernel does not
// overwrite (m >= 128 or n >= 128). 128 % 32 == 0 -> predicate is wave-uniform.
__global__ void zero_kernel(float4* __restrict__ out) {
  const int i = blockIdx.x * blockDim.x + threadIdx.x;  // one float4 per (b,m,n)
  const int mn = i & 65535;
  const int m = mn >> 8;
  const int n = mn & 255;
  if (m < 128 && n < 128) return;  // compute kernel owns this quadrant
  out[i] = make_float4(0.f, 0.f, 0.f, 0.f);
}

extern "C" void kernel_launch(void* const* d_in, const int* in_sizes, int n_in,
                              void* d_out, int out_size, void* d_ws, size_t ws_size,
                              hipStream_t stream) {
  (void)in_sizes; (void)n_in; (void)out_size; (void)ws_size;
  const float* img     = (const float*)d_in[0];  // (16,512,512,1) f32
  const float* weights = (const float*)d_in[1];  // (4,8) f32
  float* out = (float*)d_out;                    // (16,256,256,4) f32
  float* ws  = (float*)d_ws;                     // 512 floats: Ur | Ui

  // 16*256*256 float4 positions; stores only outside the computed quadrant
  zero_kernel<<<4096, 256, 0, stream>>>((float4*)out);
  build_unitary_kernel<<<1, 16, 0, stream>>>(weights, ws);
  // 262144 patches / 16 per wave = 16384 waves; 8 waves per 256-thread block
  qonv_wmma_kernel<<<2048, 256, 0, stream>>>(img, ws, out);
}